// DGProjectionBatchSparsity_64252710748190
// MI455X (gfx1250) — compile-verified
//
#include <hip/hip_runtime.h>
#include <hip/hip_bf16.h>

typedef __attribute__((ext_vector_type(2))) float v2f;
typedef __attribute__((ext_vector_type(8))) float v8f;

#define BATCH   4096
#define IN_F    512
#define OUT_F   8192
#define TOPK    204          // max(1, int(0.05 * 4096))

#define COLS_PER_WG 16
#define SEL_THREADS 512
#define KEY_STRIDE  4097     // 4096 + 1 (odd stride -> ~conflict-free LDS)

__device__ __forceinline__ unsigned sort_key(float f) {
    unsigned u = __float_as_uint(f);
    // monotone map: larger float -> larger uint
    return (u & 0x80000000u) ? ~u : (u | 0x80000000u);
}

__device__ __forceinline__ v8f wmma_f32(v2f a, v2f b, v8f c) {
    return __builtin_amdgcn_wmma_f32_16x16x4_f32(
        false, a, false, b, (short)0, c, false, false);
}

// ---------------------------------------------------------------------------
// Kernel 1: logits = x @ W^T via V_WMMA_F32_16X16X4_F32 (exact fp32).
// Workgroup tile 128(M) x 64(N); 8 waves; each wave owns a 2x2 quad of
// 16x16 tiles (4 accumulators) -> 4 WMMAs per 8 LDS dwords per K-step.
// K staged through LDS in chunks of 32 (odd stride 33: conflict-free).
// ---------------------------------------------------------------------------
__global__ __launch_bounds__(256) void dg_gemm_wmma_f32(
    const float* __restrict__ x, const float* __restrict__ W,
    float* __restrict__ out)
{
    __shared__ float xs[128][33];  // x tile [m][k]
    __shared__ float wl[64][33];   // W tile [n][k]  (B = W^T)

    const int n0   = blockIdx.x * 64;
    const int m0   = blockIdx.y * 128;
    const int t    = threadIdx.x;
    const int wave = t >> 5;
    const int lane = t & 31;
    const int lh   = lane >> 4;     // lane half: 0 or 1
    const int ll   = lane & 15;
    const int tmA  = (wave >> 1) * 32;  // first M tile row of this wave's quad
    const int tnA  = (wave & 1) * 32;   // first N tile col of this wave's quad

    v8f c00 = {}, c01 = {}, c10 = {}, c11 = {};

    for (int kc = 0; kc < IN_F; kc += 32) {
        // --- stage global -> LDS (coalesced float4 rows) ---
        float4 xv[4], wv[2];
        #pragma unroll
        for (int i = 0; i < 4; ++i) {          // 128x32 floats = 1024 float4
            const int s = t + i * 256;
            const int row = s >> 3, c4 = (s & 7) << 2;
            xv[i] = *(const float4*)(x + (size_t)(m0 + row) * IN_F + kc + c4);
        }
        #pragma unroll
        for (int i = 0; i < 2; ++i) {          // 64x32 floats = 512 float4
            const int s = t + i * 256;
            const int row = s >> 3, c4 = (s & 7) << 2;
            wv[i] = *(const float4*)(W + (size_t)(n0 + row) * IN_F + kc + c4);
        }
        __syncthreads();                       // protect previous chunk reads
        #pragma unroll
        for (int i = 0; i < 4; ++i) {
            const int s = t + i * 256;
            const int row = s >> 3, c4 = (s & 7) << 2;
            xs[row][c4 + 0] = xv[i].x; xs[row][c4 + 1] = xv[i].y;
            xs[row][c4 + 2] = xv[i].z; xs[row][c4 + 3] = xv[i].w;
        }
        #pragma unroll
        for (int i = 0; i < 2; ++i) {
            const int s = t + i * 256;
            const int row = s >> 3, c4 = (s & 7) << 2;
            wl[row][c4 + 0] = wv[i].x; wl[row][c4 + 1] = wv[i].y;
            wl[row][c4 + 2] = wv[i].z; wl[row][c4 + 3] = wv[i].w;
        }
        __syncthreads();

        // --- compute: 8 K-steps x 4 WMMAs ---
        #pragma unroll
        for (int kk = 0; kk < 32; kk += 4) {
            // ISA 7.12.2 32-bit A/B layout: lane=(K/2)*16+{M|N}, VGPR=K%2
            const int ke = kk + 2 * lh;
            v2f a0, a1, b0, b1;
            a0.x = xs[tmA + ll][ke];      a0.y = xs[tmA + ll][ke + 1];
            a1.x = xs[tmA + 16 + ll][ke]; a1.y = xs[tmA + 16 + ll][ke + 1];
            b0.x = wl[tnA + ll][ke];      b0.y = wl[tnA + ll][ke + 1];
            b1.x = wl[tnA + 16 + ll][ke]; b1.y = wl[tnA + 16 + ll][ke + 1];
            c00 = wmma_f32(a0, b0, c00);
            c01 = wmma_f32(a0, b1, c01);
            c10 = wmma_f32(a1, b0, c10);
            c11 = wmma_f32(a1, b1, c11);
        }
    }

    // C/D layout: VGPR v -> M = v + 8*lh, N = ll
    const int nB  = n0 + tnA + ll;
    const int mB0 = m0 + tmA + lh * 8;
    #pragma unroll
    for (int v = 0; v < 8; ++v) {
        out[(size_t)(mB0 + v) * OUT_F + nB]           = c00[v];
        out[(size_t)(mB0 + v) * OUT_F + nB + 16]      = c01[v];
        out[(size_t)(mB0 + 16 + v) * OUT_F + nB]      = c10[v];
        out[(size_t)(mB0 + 16 + v) * OUT_F + nB + 16] = c11[v];
    }
}

// ---------------------------------------------------------------------------
// Kernel 2: exact per-column k-th largest. 16 columns per workgroup staged
// into 256KB dynamic LDS (coalesced transpose load), one wave per column,
// 32-step bitwise binary search on count(key >= cand) >= k.
// ---------------------------------------------------------------------------
__global__ __launch_bounds__(SEL_THREADS) void dg_select_thresh(
    const float* __restrict__ logits, unsigned* __restrict__ thr)
{
    extern __shared__ unsigned keys[];  // [COLS_PER_WG][KEY_STRIDE]
    const int o0 = blockIdx.x * COLS_PER_WG;
    const int t  = threadIdx.x;

    // coalesced load: consecutive threads cover 16 consecutive columns of a row
    for (int i = t; i < BATCH * COLS_PER_WG; i += SEL_THREADS) {
        const int col = i & (COLS_PER_WG - 1);
        const int row = i >> 4;
        const float f = logits[(size_t)row * OUT_F + o0 + col];
        keys[col * KEY_STRIDE + row] = sort_key(f);
    }
    __syncthreads();

    const int wave = t >> 5;
    const int lane = t & 31;
    const unsigned* ck = keys + wave * KEY_STRIDE;

    unsigned v = 0u;
    for (int bit = 31; bit >= 0; --bit) {
        const unsigned cand = v | (1u << bit);
        int cnt = 0;
        for (int r = lane; r < BATCH; r += 32)
            cnt += (ck[r] >= cand) ? 1 : 0;
        #pragma unroll
        for (int off = 16; off > 0; off >>= 1)
            cnt += __shfl_xor(cnt, off, 32);
        if (cnt >= TOPK) v = cand;      // uniform across wave
    }
    if (lane == 0) thr[o0 + wave] = v;  // exact k-th largest key
}

// ---------------------------------------------------------------------------
// Kernel 3: in-place mask: out = (key(logit) >= thr[col]) ? 1 : 0
// ---------------------------------------------------------------------------
__global__ __launch_bounds__(256) void dg_write_mask(
    float* __restrict__ inout, const unsigned* __restrict__ thr)
{
    const size_t idx = ((size_t)blockIdx.x * 256 + threadIdx.x) * 4;
    const int o = (int)(idx & (OUT_F - 1));   // 4 elems never cross a row
    float4 lv = *(float4*)(inout + idx);
    float4 r;
    r.x = (sort_key(lv.x) >= thr[o + 0]) ? 1.0f : 0.0f;
    r.y = (sort_key(lv.y) >= thr[o + 1]) ? 1.0f : 0.0f;
    r.z = (sort_key(lv.z) >= thr[o + 2]) ? 1.0f : 0.0f;
    r.w = (sort_key(lv.w) >= thr[o + 3]) ? 1.0f : 0.0f;
    *(float4*)(inout + idx) = r;
}

// ---------------------------------------------------------------------------
extern "C" void kernel_launch(void* const* d_in, const int* in_sizes, int n_in,
                              void* d_out, int out_size, void* d_ws, size_t ws_size,
                              hipStream_t stream) {
    (void)in_sizes; (void)n_in; (void)out_size; (void)ws_size;
    const float* x = (const float*)d_in[0];
    const float* W = (const float*)d_in[1];
    // d_in[2] = bias: forward output (binary top-k mask) is invariant to it.
    float*    out = (float*)d_out;          // used as logits scratch, then mask
    unsigned* thr = (unsigned*)d_ws;        // 8192 * 4B thresholds

    dim3 g1(OUT_F / 64, BATCH / 128);
    dg_gemm_wmma_f32<<<g1, 256, 0, stream>>>(x, W, out);

    const size_t lds2 = (size_t)COLS_PER_WG * KEY_STRIDE * sizeof(unsigned); // ~256KB
    dg_select_thresh<<<OUT_F / COLS_PER_WG, SEL_THREADS, lds2, stream>>>(out, thr);

    const size_t nvec = (size_t)BATCH * OUT_F / 4;
    dg_write_mask<<<(unsigned)(nvec / 256), 256, 0, stream>>>(out, thr);
}